// diffeomorphicTransformer3D_47175920779660
// MI455X (gfx1250) — compile-verified
//
#include <hip/hip_runtime.h>

// Trilinear 3D warp (grid sample) for im[B,X,Y,Z,C], defgrid[B,X,Y,Z,3].
// Memory-bound gather kernel: im (65.5 MB) fits in MI455X's 192 MB L2, so we
// stream defgrid/out with non-temporal policy (gfx1250 TH=NT) to keep im
// resident, and fetch each corner's 2 channels as one global_load_b64.

typedef float v2f __attribute__((ext_vector_type(2)));   // native vector: OK for NT builtins

namespace {
constexpr int Bc = 2, Xc = 160, Yc = 160, Zc = 160, Cc = 2;
constexpr int VOL  = Xc * Yc * Zc;   // 4,096,000 voxels per batch
constexpr int NPTS = Bc * VOL;       // 8,192,000 output points
constexpr int DIM3 = Zc;             // y-stride in voxels
constexpr int DIM2 = Yc * Zc;        // x-stride in voxels
constexpr int BLOCK = 256;           // 8 wave32 waves per block
}

__global__ __launch_bounds__(BLOCK) void diffeo_trilerp3d_kernel(
    const float* __restrict__ im,
    const float* __restrict__ defgrid,
    float* __restrict__ out)
{
    const int i = blockIdx.x * BLOCK + threadIdx.x;
    if (i >= NPTS) return;

    // ---- streamed coordinate read (non-temporal: touched exactly once) ----
    const float* g = defgrid + 3ll * i;
    const float gx = __builtin_nontemporal_load(g + 0);
    const float gy = __builtin_nontemporal_load(g + 1);
    const float gz = __builtin_nontemporal_load(g + 2);

    const int x0u = (int)floorf(gx);
    const int y0u = (int)floorf(gy);
    const int z0u = (int)floorf(gz);

    const int x0 = min(max(x0u,     0), Xc - 1);
    const int x1 = min(max(x0u + 1, 0), Xc - 1);
    const int y0 = min(max(y0u,     0), Yc - 1);
    const int y1 = min(max(y0u + 1, 0), Yc - 1);
    const int z0 = min(max(z0u,     0), Zc - 1);
    const int z1 = min(max(z0u + 1, 0), Zc - 1);

    // fractional weights use the CLIPPED lower corner (matches reference)
    const float xd = gx - (float)x0;
    const float yd = gy - (float)y0;
    const float zd = gz - (float)z0;
    const float wx0 = 1.0f - xd;
    const float wy0 = 1.0f - yd;
    const float wz0 = 1.0f - zd;

    // batch base: B == 2, so a compare beats an integer divide
    const int base = (i >= VOL) ? VOL : 0;

    // ---- 8 corner gathers, 2 contiguous channels each -> global_load_b64 ----
    const v2f* __restrict__ imv = (const v2f*)im;
    const int r00 = base + x0 * DIM2 + y0 * DIM3;
    const int r01 = base + x0 * DIM2 + y1 * DIM3;
    const int r10 = base + x1 * DIM2 + y0 * DIM3;
    const int r11 = base + x1 * DIM2 + y1 * DIM3;

    const v2f Ia = imv[r00 + z0];   // (x0,y0,z0)
    const v2f Ib = imv[r00 + z1];   // (x0,y0,z1)
    const v2f Ic = imv[r01 + z0];   // (x0,y1,z0)
    const v2f Id = imv[r01 + z1];   // (x0,y1,z1)
    const v2f Ie = imv[r10 + z0];   // (x1,y0,z0)
    const v2f If = imv[r10 + z1];   // (x1,y0,z1)
    const v2f Ig = imv[r11 + z0];   // (x1,y1,z0)
    const v2f Ih = imv[r11 + z1];   // (x1,y1,z1)

    // ---- trilinear blend, both channels (same op order as reference) ----
    // native vectors support elementwise arithmetic directly
    const v2f Cae = Ia * wx0 + Ie * xd;
    const v2f Cbf = Ib * wx0 + If * xd;
    const v2f Ccg = Ic * wx0 + Ig * xd;
    const v2f Cdh = Id * wx0 + Ih * xd;

    const v2f Caecg = Cae * wy0 + Ccg * yd;
    const v2f Cbfdh = Cbf * wy0 + Cdh * yd;

    const v2f res = Caecg * wz0 + Cbfdh * zd;

    // ---- streamed result write (non-temporal b64 store) ----
    __builtin_nontemporal_store(res, (v2f*)out + i);
}

extern "C" void kernel_launch(void* const* d_in, const int* in_sizes, int n_in,
                              void* d_out, int out_size, void* d_ws, size_t ws_size,
                              hipStream_t stream) {
    (void)in_sizes; (void)n_in; (void)d_ws; (void)ws_size; (void)out_size;
    const float* im      = (const float*)d_in[0];   // [B,X,Y,Z,C] fp32
    const float* defgrid = (const float*)d_in[1];   // [B,X,Y,Z,3] fp32
    float*       out     = (float*)d_out;           // [B,X,Y,Z,C] fp32

    const int grid = (NPTS + BLOCK - 1) / BLOCK;    // 32,000 blocks
    diffeo_trilerp3d_kernel<<<grid, BLOCK, 0, stream>>>(im, defgrid, out);
}